// SegRNN_50328426774783
// MI455X (gfx1250) — compile-verified
//
#include <hip/hip_runtime.h>
#include <hip/hip_bf16.h>
#include <math.h>

typedef float v2f __attribute__((ext_vector_type(2)));
typedef float v8f __attribute__((ext_vector_type(8)));

#define N_    256
#define B_    16
#define K_    64
#define X_    128
#define S_    128
#define TAG_  32
#define DUR_  16
#define Y_    16
#define L_    16
#define SMAX_ 32
#define NEGV  (-1.0e30f)

#define HSTR 132   // LDS stride for h (128 cols): 132%64==4 -> conflict-free b64 loads
#define GSTR 520   // LDS stride for gate tile (512 cols)
#define ASTR 260   // LDS stride for 256-col A tile
#define PSTR 132   // LDS stride for 128-col pre tile

__device__ __forceinline__ float sigf(float x) { return 1.0f / (1.0f + expf(-x)); }

__device__ __forceinline__ v8f wmma4(v2f a, v2f b, v8f c) {
  // V_WMMA_F32_16X16X4_F32: D = A(16x4) * B(4x16) + C(16x16), fp32 throughout
  return __builtin_amdgcn_wmma_f32_16x16x4_f32(false, a, false, b, (short)0, c,
                                               false, false);
}

// ---------------------------------------------------------------------------
// Generic WMMA GEMM: C[M,N] = A[M,K] @ W[N,K]^T + b1 + b2
// grid = (N/64, M/16), block = 128 (4 waves, 1 16x16 N-tile per wave)
// ---------------------------------------------------------------------------
__global__ __launch_bounds__(128) void gemm_wmma_kernel(
    const float* __restrict__ A, int lda,
    const float* __restrict__ W, int ldw,
    const float* __restrict__ b1, const float* __restrict__ b2,
    float* __restrict__ C, int ldc, int K)
{
  int wave  = threadIdx.x >> 5;
  int lane  = threadIdx.x & 31;
  int m0    = blockIdx.y * 16;
  int n0    = blockIdx.x * 64 + wave * 16;
  int mlane = lane & 15;
  int dk    = (lane >> 4) * 2;   // K sub-offset for this half-wave

  const float* Ap = A + (m0 + mlane) * lda + dk;
  const float* Wp = W + (n0 + mlane) * ldw + dk;

  v8f acc = {0.f, 0.f, 0.f, 0.f, 0.f, 0.f, 0.f, 0.f};
  for (int k0 = 0; k0 < K; k0 += 4) {
    v2f a = *(const v2f*)(Ap + k0);
    v2f b = *(const v2f*)(Wp + k0);
    acc = wmma4(a, b, acc);
  }

  int n = n0 + mlane;
  float bias = (b1 ? b1[n] : 0.0f) + (b2 ? b2[n] : 0.0f);
  int mbase = m0 + ((lane >> 4) << 3);
#pragma unroll
  for (int r = 0; r < 8; ++r)
    C[(mbase + r) * ldc + n] = acc[r] + bias;
}

// ---------------------------------------------------------------------------
// ctx bidirectional LSTM: grid.x = dir (0 fwd, 1 bwd), block = 256 (8 waves)
// P = precomputed input projections (N*B, 512); sequential over N steps.
// Per step: g = h @ Wh^T via WMMA (h kept in LDS), then gate nonlinearity.
// ---------------------------------------------------------------------------
__global__ __launch_bounds__(256) void ctx_lstm_kernel(
    const float* __restrict__ Pf, const float* __restrict__ Pb,
    const float* __restrict__ Whf, const float* __restrict__ Whb,
    const float* __restrict__ h0f, const float* __restrict__ c0f,
    const float* __restrict__ h0b, const float* __restrict__ c0b,
    float* __restrict__ xcribe)
{
  __shared__ float hs[16 * HSTR];
  __shared__ float gs[16 * GSTR];

  int dir = blockIdx.x;
  const float* P  = dir ? Pb  : Pf;
  const float* Wh = dir ? Whb : Whf;
  const float* h0 = dir ? h0b : h0f;
  const float* c0 = dir ? c0b : c0f;

  int t = threadIdx.x;
  int wave = t >> 5, lane = t & 31;
  int mlane = lane & 15, dk = (lane >> 4) * 2;
  int be = t >> 4;               // batch row handled in elementwise phase
  int j0 = (t & 15) * 8;         // 8 consecutive hidden units

  float creg[8];
#pragma unroll
  for (int i = 0; i < 8; ++i) {
    hs[be * HSTR + j0 + i] = h0[j0 + i];
    creg[i] = c0[j0 + i];
  }
  __syncthreads();

  for (int s = 0; s < N_; ++s) {
    int n = dir ? (N_ - 1 - s) : s;

    // --- recurrent GEMM: wave owns 4 N-tiles (64 gate columns) ---
    v8f acc[4];
#pragma unroll
    for (int ti = 0; ti < 4; ++ti) acc[ti] = (v8f){0.f,0.f,0.f,0.f,0.f,0.f,0.f,0.f};
    const float* hp = &hs[mlane * HSTR + dk];
    for (int k0 = 0; k0 < X_; k0 += 4) {
      v2f a = *(const v2f*)(hp + k0);
#pragma unroll
      for (int ti = 0; ti < 4; ++ti) {
        int nn = wave * 64 + ti * 16 + mlane;
        v2f b = *(const v2f*)(Wh + nn * X_ + dk + k0);
        acc[ti] = wmma4(a, b, acc[ti]);
      }
    }
    int mb = (lane >> 4) << 3;
#pragma unroll
    for (int ti = 0; ti < 4; ++ti) {
      int nn = wave * 64 + ti * 16 + mlane;
#pragma unroll
      for (int r = 0; r < 8; ++r)
        gs[(mb + r) * GSTR + nn] = acc[ti][r];
    }
    __syncthreads();   // all h reads done; gate tile complete

    // --- gates + state update ---
    const float* Pr = P + (n * B_ + be) * 512;
#pragma unroll
    for (int i = 0; i < 8; ++i) {
      int j = j0 + i;
      float gi = gs[be * GSTR + j]       + Pr[j];
      float gf = gs[be * GSTR + 128 + j] + Pr[128 + j];
      float gg = gs[be * GSTR + 256 + j] + Pr[256 + j];
      float go = gs[be * GSTR + 384 + j] + Pr[384 + j];
      float c  = sigf(gf) * creg[i] + sigf(gi) * tanhf(gg);
      creg[i]  = c;
      float h  = sigf(go) * tanhf(c);
      hs[be * HSTR + j] = h;
      xcribe[(n * B_ + be) * 256 + dir * 128 + j] = h;
    }
    __syncthreads();   // h ready for next step's GEMM
  }
}

// ---------------------------------------------------------------------------
// segment LSTM: grid.x = chain n (256 blocks), 16 steps. sgn=+1 fwd / -1 bwd.
// Input projections gathered as P[clip(n + sgn*l)].
// ---------------------------------------------------------------------------
__global__ __launch_bounds__(256) void seg_lstm_kernel(
    const float* __restrict__ P, const float* __restrict__ Wh,
    const float* __restrict__ h0, const float* __restrict__ c0,
    float* __restrict__ out, int sgn)
{
  __shared__ float hs[16 * HSTR];
  __shared__ float gs[16 * GSTR];

  int nchain = blockIdx.x;
  int t = threadIdx.x;
  int wave = t >> 5, lane = t & 31;
  int mlane = lane & 15, dk = (lane >> 4) * 2;
  int be = t >> 4;
  int j0 = (t & 15) * 8;

  float creg[8];
#pragma unroll
  for (int i = 0; i < 8; ++i) {
    hs[be * HSTR + j0 + i] = h0[j0 + i];
    creg[i] = c0[j0 + i];
  }
  __syncthreads();

  for (int l = 0; l < L_; ++l) {
    int xi = nchain + sgn * l;
    xi = xi < 0 ? 0 : (xi > N_ - 1 ? N_ - 1 : xi);

    v8f acc[4];
#pragma unroll
    for (int ti = 0; ti < 4; ++ti) acc[ti] = (v8f){0.f,0.f,0.f,0.f,0.f,0.f,0.f,0.f};
    const float* hp = &hs[mlane * HSTR + dk];
    for (int k0 = 0; k0 < S_; k0 += 4) {
      v2f a = *(const v2f*)(hp + k0);
#pragma unroll
      for (int ti = 0; ti < 4; ++ti) {
        int nn = wave * 64 + ti * 16 + mlane;
        v2f b = *(const v2f*)(Wh + nn * S_ + dk + k0);
        acc[ti] = wmma4(a, b, acc[ti]);
      }
    }
    int mb = (lane >> 4) << 3;
#pragma unroll
    for (int ti = 0; ti < 4; ++ti) {
      int nn = wave * 64 + ti * 16 + mlane;
#pragma unroll
      for (int r = 0; r < 8; ++r)
        gs[(mb + r) * GSTR + nn] = acc[ti][r];
    }
    __syncthreads();

    const float* Pr = P + (xi * B_ + be) * 512;
#pragma unroll
    for (int i = 0; i < 8; ++i) {
      int j = j0 + i;
      float gi = gs[be * GSTR + j]       + Pr[j];
      float gf = gs[be * GSTR + 128 + j] + Pr[128 + j];
      float gg = gs[be * GSTR + 256 + j] + Pr[256 + j];
      float go = gs[be * GSTR + 384 + j] + Pr[384 + j];
      float c  = sigf(gf) * creg[i] + sigf(gi) * tanhf(gg);
      creg[i]  = c;
      float h  = sigf(go) * tanhf(c);
      hs[be * HSTR + j] = h;
      out[((nchain * L_ + l) * B_ + be) * S_ + j] = h;
    }
    __syncthreads();
  }
}

// ---------------------------------------------------------------------------
// score + logsumexp over Y: grid = (L, N) -> one (d,e) tile per block.
// Gather h_cat rows -> WMMA pre = h_cat @ Vh^T (+Vb +zpart) -> tanh scores ->
// LSE over y. Invalid (e<d) blocks just write NEG.
// ---------------------------------------------------------------------------
__global__ __launch_bounds__(256) void score_lse_kernel(
    const float* __restrict__ fseg, const float* __restrict__ bseg,
    const float* __restrict__ Vw, const float* __restrict__ Vb,
    const float* __restrict__ ypart, const float* __restrict__ zpart,
    const float* __restrict__ Ww, const float* __restrict__ Wb,
    float* __restrict__ lse_y)
{
  int d = blockIdx.x;
  int e = blockIdx.y;
  int t = threadIdx.x;

  if (e < d) {                    // uniform per block: no WMMA executed
    if (t < B_) lse_y[(e * L_ + d) * B_ + t] = NEGV;
    return;
  }

  __shared__ float As[16 * ASTR];
  __shared__ float ps[16 * PSTR];
  __shared__ float sc[Y_ * 17];

  int e2 = e - d;                 // fwd gather index (already >= 0)
  for (int i = t; i < 16 * 256; i += 256) {
    int b = i >> 8, h = i & 255;
    float v = (h < 128) ? fseg[((e2 * L_ + d) * B_ + b) * S_ + h]
                        : bseg[((e  * L_ + d) * B_ + b) * S_ + (h - 128)];
    As[b * ASTR + h] = v;
  }
  __syncthreads();

  // pre[16x128] = A @ Vh^T : 8 waves, one 16x16 tile each
  int wave = t >> 5, lane = t & 31;
  int mlane = lane & 15, dk = (lane >> 4) * 2;
  v8f acc = {0.f,0.f,0.f,0.f,0.f,0.f,0.f,0.f};
  const float* ap = &As[mlane * ASTR + dk];
  const float* wp = Vw + (wave * 16 + mlane) * 320 + dk;
  for (int k0 = 0; k0 < 256; k0 += 4) {
    v2f a = *(const v2f*)(ap + k0);
    v2f b = *(const v2f*)(wp + k0);
    acc = wmma4(a, b, acc);
  }
  int n = wave * 16 + mlane;
  float add = Vb[n] + zpart[d * 128 + n];
  int mb = (lane >> 4) << 3;
#pragma unroll
  for (int r = 0; r < 8; ++r)
    ps[(mb + r) * PSTR + n] = acc[r] + add;
  __syncthreads();

  // score[y][b] = Wb + sum_o tanh(pre[b][o] + ypart[y][o]) * Ww[o]
  int y = t >> 4, b = t & 15;
  const float* pr = &ps[b * PSTR];
  const float* yp = ypart + y * 128;
  float s = 0.f;
  for (int o = 0; o < 128; ++o)
    s += tanhf(pr[o] + yp[o]) * Ww[o];
  sc[y * 17 + b] = s + Wb[0];
  __syncthreads();

  if (t < B_) {
    float m = sc[t];
    for (int yy = 1; yy < Y_; ++yy) m = fmaxf(m, sc[yy * 17 + t]);
    float su = 0.f;
    for (int yy = 0; yy < Y_; ++yy) su += expf(sc[yy * 17 + t] - m);
    lse_y[(e * L_ + d) * B_ + t] = m + logf(su);
  }
}

// ---------------------------------------------------------------------------
// semi-CRF DP scan: single block, 256 threads = (l,b). Sequential over N.
// ---------------------------------------------------------------------------
__global__ __launch_bounds__(256) void dp_kernel(
    const float* __restrict__ lse_y, float* __restrict__ logZ)
{
  __shared__ float buf[L_ * B_];
  __shared__ float tmp[L_ * B_];
  __shared__ float nw[B_];

  int t = threadIdx.x;
  int l = t >> 4, b = t & 15;
  buf[t] = (l == 0) ? 0.0f : NEGV;
  __syncthreads();

  for (int e = 0; e < N_; ++e) {
    tmp[t] = buf[t] + lse_y[(e * L_ + l) * B_ + b];
    __syncthreads();
    if (l == 0) {
      float m = tmp[b];
      for (int ll = 1; ll < L_; ++ll) m = fmaxf(m, tmp[ll * 16 + b]);
      float su = 0.f;
      for (int ll = 0; ll < L_; ++ll) su += expf(tmp[ll * 16 + b] - m);
      nw[b] = m + logf(su);
    }
    __syncthreads();
    float old = (l == 0) ? nw[b] : buf[(l - 1) * 16 + b];
    __syncthreads();
    buf[t] = old;                 // shift: buf = [new, buf[:-1]]
    __syncthreads();
  }
  if (t == 0) {
    float s = 0.f;
    for (int bb = 0; bb < B_; ++bb) s += nw[bb];
    *logZ = s;
  }
}

// ---------------------------------------------------------------------------
// gold path score: grid = B_, block = 128 (one thread per output unit o)
// ---------------------------------------------------------------------------
__global__ __launch_bounds__(128) void gold_kernel(
    const float* __restrict__ fseg, const float* __restrict__ bseg,
    const float* __restrict__ Y_enc, const float* __restrict__ Z_enc,
    const float* __restrict__ Vw, const float* __restrict__ Vb,
    const float* __restrict__ Ww, const float* __restrict__ Wb,
    const int* __restrict__ tags, const int* __restrict__ lengths,
    float* __restrict__ gold)
{
  int b = blockIdx.x;
  int t = threadIdx.x;
  __shared__ float senc[320];
  __shared__ float red[128];
  __shared__ int   st[SMAX_];
  __shared__ float bsum;

  if (t == 0) {
    int acc = 0;
    for (int s = 0; s < SMAX_; ++s) { st[s] = acc; acc += lengths[b * SMAX_ + s]; }
    bsum = 0.f;
  }
  __syncthreads();

  for (int s = 0; s < SMAX_; ++s) {
    int len = lengths[b * SMAX_ + s];
    int s0  = st[s];
    int en  = s0 + len - 1;
    bool valid = (len > 0) && (s0 + len <= N_);
    int d_i = min(max(len - 1, 0), L_ - 1);
    int s_i = min(max(s0, 0), N_ - 1);
    int e_i = min(max(en, 0), N_ - 1);
    int tag = tags[b * SMAX_ + s];

    for (int i = t; i < 320; i += 128) {
      float v;
      if (i < 128)      v = fseg[((s_i * L_ + d_i) * B_ + b) * S_ + i];
      else if (i < 256) v = bseg[((e_i * L_ + d_i) * B_ + b) * S_ + (i - 128)];
      else if (i < 288) v = Y_enc[tag * TAG_ + (i - 256)];
      else              v = Z_enc[d_i * DUR_ + (i - 288)];
      senc[i] = v;
    }
    __syncthreads();

    float dot = Vb[t];
    const float* vr = Vw + t * 320;
    for (int k = 0; k < 320; ++k) dot += vr[k] * senc[k];
    red[t] = tanhf(dot) * Ww[t];
    __syncthreads();
    for (int off = 64; off > 0; off >>= 1) {
      if (t < off) red[t] += red[t + off];
      __syncthreads();
    }
    if (t == 0 && valid) bsum += red[0] + Wb[0];
    __syncthreads();
  }
  if (t == 0) atomicAdd(gold, bsum);
}

__global__ void init_kernel(float* scal) {
  if (threadIdx.x < 8) scal[threadIdx.x] = 0.f;
}

__global__ void finalize_kernel(const float* __restrict__ scal,
                                float* __restrict__ out) {
  if (threadIdx.x == 0) out[0] = scal[0] - scal[1];  // logZ - gold
}

// ---------------------------------------------------------------------------
extern "C" void kernel_launch(void* const* d_in, const int* in_sizes, int n_in,
                              void* d_out, int out_size, void* d_ws, size_t ws_size,
                              hipStream_t stream) {
  const float* data    = (const float*)d_in[0];
  const float* cfWi = (const float*)d_in[1];  const float* cfWh = (const float*)d_in[2];
  const float* cfbi = (const float*)d_in[3];  const float* cfbh = (const float*)d_in[4];
  const float* cfh0 = (const float*)d_in[5];  const float* cfc0 = (const float*)d_in[6];
  const float* cbWi = (const float*)d_in[7];  const float* cbWh = (const float*)d_in[8];
  const float* cbbi = (const float*)d_in[9];  const float* cbbh = (const float*)d_in[10];
  const float* cbh0 = (const float*)d_in[11]; const float* cbc0 = (const float*)d_in[12];
  const float* sfWi = (const float*)d_in[13]; const float* sfWh = (const float*)d_in[14];
  const float* sfbi = (const float*)d_in[15]; const float* sfbh = (const float*)d_in[16];
  const float* sfh0 = (const float*)d_in[17]; const float* sfc0 = (const float*)d_in[18];
  const float* sbWi = (const float*)d_in[19]; const float* sbWh = (const float*)d_in[20];
  const float* sbbi = (const float*)d_in[21]; const float* sbbh = (const float*)d_in[22];
  const float* sbh0 = (const float*)d_in[23]; const float* sbc0 = (const float*)d_in[24];
  const float* Y_enc = (const float*)d_in[25];
  const float* Z_enc = (const float*)d_in[26];
  const float* Vw    = (const float*)d_in[27];
  const float* Vb    = (const float*)d_in[28];
  const float* Ww    = (const float*)d_in[29];
  const float* Wb    = (const float*)d_in[30];
  const int*   tags    = (const int*)d_in[31];
  const int*   lengths = (const int*)d_in[32];

  // workspace layout (floats); ctx-proj buffers are reused for seg-projs
  float* ws      = (float*)d_ws;
  float* xcribe  = ws;                       // 4096*256   = 1,048,576
  float* Pcf     = xcribe + 1048576;         // 4096*512   = 2,097,152
  float* Pcb     = Pcf    + 2097152;         // 2,097,152
  float* Psf     = Pcf;                      // alias (ctx projs dead by then)
  float* Psb     = Pcb;                      // alias
  float* fseg    = Pcb    + 2097152;         // 256*16*16*128 = 8,388,608
  float* bseg    = fseg   + 8388608;         // 8,388,608
  float* ypart   = bseg   + 8388608;         // 16*128
  float* zpart   = ypart  + 2048;            // 16*128
  float* lse     = zpart  + 2048;            // 256*16*16 = 65,536
  float* scal    = lse    + 65536;           // [0]=logZ [1]=gold

  init_kernel<<<1, 32, 0, stream>>>(scal);

  // ctx input projections: (4096,64) @ (512,64)^T + bi + bh
  dim3 gP(512 / 64, (N_ * B_) / 16);
  gemm_wmma_kernel<<<gP, 128, 0, stream>>>(data, K_, cfWi, K_, cfbi, cfbh, Pcf, 512, K_);
  gemm_wmma_kernel<<<gP, 128, 0, stream>>>(data, K_, cbWi, K_, cbbi, cbbh, Pcb, 512, K_);

  // bidirectional context LSTM -> xcribe (N,B,256)
  ctx_lstm_kernel<<<2, 256, 0, stream>>>(Pcf, Pcb, cfWh, cbWh,
                                         cfh0, cfc0, cbh0, cbc0, xcribe);

  // seg input projections: (4096,256) @ (512,256)^T + bi + bh
  gemm_wmma_kernel<<<gP, 128, 0, stream>>>(xcribe, 256, sfWi, 256, sfbi, sfbh, Psf, 512, 256);
  gemm_wmma_kernel<<<gP, 128, 0, stream>>>(xcribe, 256, sbWi, 256, sbbi, sbbh, Psb, 512, 256);

  // segment LSTMs: 256 chains x 16 steps
  seg_lstm_kernel<<<N_, 256, 0, stream>>>(Psf, sfWh, sfh0, sfc0, fseg, +1);
  seg_lstm_kernel<<<N_, 256, 0, stream>>>(Psb, sbWh, sbh0, sbc0, bseg, -1);

  // y/z partial projections: Y_enc @ Vy^T, Z_enc @ Vz^T (Vy/Vz are V_w slices)
  dim3 gYZ(128 / 64, 1);
  gemm_wmma_kernel<<<gYZ, 128, 0, stream>>>(Y_enc, TAG_, Vw + 256, 320,
                                            nullptr, nullptr, ypart, 128, TAG_);
  gemm_wmma_kernel<<<gYZ, 128, 0, stream>>>(Z_enc, DUR_, Vw + 288, 320,
                                            nullptr, nullptr, zpart, 128, DUR_);

  // fused gather + pre-GEMM + tanh scores + LSE over Y
  dim3 gS(L_, N_);
  score_lse_kernel<<<gS, 256, 0, stream>>>(fseg, bseg, Vw, Vb, ypart, zpart,
                                           Ww, Wb, lse);

  dp_kernel<<<1, 256, 0, stream>>>(lse, scal);           // -> scal[0] = logZ
  gold_kernel<<<B_, 128, 0, stream>>>(fseg, bseg, Y_enc, Z_enc, Vw, Vb,
                                      Ww, Wb, tags, lengths, scal + 1);
  finalize_kernel<<<1, 32, 0, stream>>>(scal, (float*)d_out);
}